// KASR_66005057405539
// MI455X (gfx1250) — compile-verified
//
#include <hip/hip_runtime.h>
#include <hip/hip_bf16.h>

// ---------------------------------------------------------------------------
// CDNA5 fp32 WMMA: D(16x16,f32) = A(16x4,f32) x B(4x16,f32) + C
// A layout: lane L (0-15): M=L, regs = K{k0,k0+1}; lanes 16-31: M=L-16, K{k0+2,k0+3}
// B layout: lane L (0-15): N=L, regs = K{k0,k0+1}; lanes 16-31: N=L-16, K{k0+2,k0+3}
// C/D layout: reg r: lanes 0-15 -> M=r, N=lane; lanes 16-31 -> M=r+8, N=lane-16
// ---------------------------------------------------------------------------
typedef float v2f __attribute__((ext_vector_type(2)));
typedef float v8f __attribute__((ext_vector_type(8)));

__device__ __forceinline__ v8f wmma4(v2f a, v2f b, v8f c) {
    return __builtin_amdgcn_wmma_f32_16x16x4_f32(
        /*neg_a=*/false, a, /*neg_b=*/false, b,
        /*c_mod=*/(short)0, c, /*reuse_a=*/false, /*reuse_b=*/false);
}

// --- CDNA5 async global->LDS copy (ASYNCcnt-tracked, no VGPR round trip) ---
// LDS generic pointers: aperture is 4GB aligned, addr[31:0] is the LDS offset.
__device__ __forceinline__ unsigned to_lds(const void* p) {
    return (unsigned)(unsigned long long)p;
}
__device__ __forceinline__ void async_copy16(unsigned lds_byte_off, const void* gaddr) {
    asm volatile("global_load_async_to_lds_b128 %0, %1, off"
                 :: "v"(lds_byte_off), "v"(gaddr) : "memory");
}
__device__ __forceinline__ void wait_async() {
    asm volatile("s_wait_asynccnt 0x0" ::: "memory");
}

#define B_    256
#define S_    50
#define NB_   8
#define D_    128
#define NI_   100000
#define NR_   200

__device__ __forceinline__ float sigmoidf_(float x) { return 1.0f / (1.0f + __expf(-x)); }

// ---------------------------------------------------------------------------
// Generic Y[M,N] = X[M,128] @ W[128,N] + bias.  grid=(M/16, N/64), block=128.
// X tile (16x128) and W tile (128x64) staged in LDS via async b128 copies;
// inner loop is pure LDS -> WMMA.
// ---------------------------------------------------------------------------
__global__ void gemm_k128(const float* __restrict__ X, const float* __restrict__ W,
                          const float* __restrict__ bias, float* __restrict__ Y,
                          int N) {
    __shared__ __align__(16) float sX[16 * 128];
    __shared__ __align__(16) float sW[128 * 64];
    const int tid  = threadIdx.x;
    const int m0   = blockIdx.x * 16;
    const int nblk = blockIdx.y * 64;

    // stage X tile: 512 float4 (contiguous rows)
    {
        const unsigned xb = to_lds(sX);
        const float4* X4 = (const float4*)(X + (size_t)m0 * 128);
        #pragma unroll
        for (int i = tid; i < 512; i += 128)
            async_copy16(xb + (unsigned)i * 16u, X4 + i);
    }
    // stage W tile: rows 0..127, cols nblk..nblk+63  (16 float4 per row)
    {
        const unsigned wb = to_lds(sW);
        #pragma unroll
        for (int i = tid; i < 2048; i += 128) {
            const int row = i >> 4, c4 = i & 15;
            async_copy16(wb + (unsigned)i * 16u,
                         (const float4*)(W + (size_t)row * N + nblk) + c4);
        }
    }
    wait_async();
    __syncthreads();

    const int wave = tid >> 5, lane = tid & 31;
    const int lm = lane & 15, hi = lane >> 4;
    const int nloc = wave * 16;

    v8f acc = {};
    #pragma unroll 8
    for (int k0 = 0; k0 < 128; k0 += 4) {
        v2f a, b;
        a.x = sX[lm * 128 + k0 + 2 * hi + 0];
        a.y = sX[lm * 128 + k0 + 2 * hi + 1];
        b.x = sW[(k0 + 2 * hi + 0) * 64 + nloc + lm];
        b.y = sW[(k0 + 2 * hi + 1) * 64 + nloc + lm];
        acc = wmma4(a, b, acc);
    }
    const int n = nblk + nloc + lm;
    const float bv = bias ? bias[n] : 0.0f;
    #pragma unroll
    for (int r = 0; r < 8; ++r) {
        const int m = m0 + r + 8 * hi;
        Y[(size_t)m * N + n] = acc[r] + bv;
    }
}

// ---------------------------------------------------------------------------
// Level-0 fused gather + attention + aggregate.
// grid = B*S*NB (one block per hop-1 node), block = 128 (thread = feature d)
// X0[node,d] = self[d] + sum_k softmax_k(<self*rel_k*nb_k, Wa>+ba) * nb_k[d]
// ---------------------------------------------------------------------------
__global__ void agg_level0(const int* __restrict__ h_iids, const int* __restrict__ adjE,
                           const int* __restrict__ adjR, const float* __restrict__ item,
                           const float* __restrict__ rel, const float* __restrict__ Wa,
                           const float* __restrict__ ba, float* __restrict__ X0) {
    const int node = blockIdx.x;               // b*400 + m,  m = s*8 + j
    const int b = node / (S_ * NB_);
    const int m = node % (S_ * NB_);
    const int s = m / NB_, j = m % NB_;
    const int d = threadIdx.x;

    __shared__ float s_self[128];
    __shared__ float s_nb[NB_][128];
    __shared__ float s_att[NB_];
    __shared__ float red[128];

    const int e0 = h_iids[b * S_ + s];
    const int e1 = adjE[(size_t)e0 * NB_ + j];
    s_self[d] = item[(size_t)e1 * D_ + d];
    __syncthreads();
    const float sw = s_self[d] * Wa[d];

    for (int k = 0; k < NB_; ++k) {
        const int e2 = adjE[(size_t)e1 * NB_ + k];
        const int r  = adjR[(size_t)e1 * NB_ + k];
        const float nb = item[(size_t)e2 * D_ + d];
        const float rl = rel[(size_t)r * D_ + d];
        s_nb[k][d] = nb;
        red[d] = sw * rl * nb;
        __syncthreads();
        for (int off = 64; off > 0; off >>= 1) {
            if (d < off) red[d] += red[d + off];
            __syncthreads();
        }
        if (d == 0) s_att[k] = red[0] + ba[0];
        __syncthreads();
    }
    float mx = -3.4e38f;
    #pragma unroll
    for (int k = 0; k < NB_; ++k) mx = fmaxf(mx, s_att[k]);
    float al[NB_], sum = 0.0f;
    #pragma unroll
    for (int k = 0; k < NB_; ++k) { al[k] = __expf(s_att[k] - mx); sum += al[k]; }
    const float inv = 1.0f / sum;
    float agg = 0.0f;
    #pragma unroll
    for (int k = 0; k < NB_; ++k) agg += al[k] * inv * s_nb[k][d];
    X0[(size_t)node * D_ + d] = s_self[d] + agg;
}

// ---------------------------------------------------------------------------
// Level-1 attention + aggregate.  grid = B*S, block = 128
// self = item_emb[h_iids], neighbors = neib0[b, s*8+k, :]
// ---------------------------------------------------------------------------
__global__ void agg_level1(const int* __restrict__ h_iids, const int* __restrict__ adjR,
                           const float* __restrict__ item, const float* __restrict__ rel,
                           const float* __restrict__ neib0, const float* __restrict__ Wa,
                           const float* __restrict__ ba, float* __restrict__ X1) {
    const int node = blockIdx.x;               // b*50 + s
    const int b = node / S_, s = node % S_;
    const int d = threadIdx.x;

    __shared__ float s_self[128];
    __shared__ float s_nb[NB_][128];
    __shared__ float s_att[NB_];
    __shared__ float red[128];

    const int e0 = h_iids[b * S_ + s];
    s_self[d] = item[(size_t)e0 * D_ + d];
    __syncthreads();
    const float sw = s_self[d] * Wa[d];

    for (int k = 0; k < NB_; ++k) {
        const int r = adjR[(size_t)e0 * NB_ + k];
        const float nb = neib0[((size_t)b * (S_ * NB_) + s * NB_ + k) * D_ + d];
        const float rl = rel[(size_t)r * D_ + d];
        s_nb[k][d] = nb;
        red[d] = sw * rl * nb;
        __syncthreads();
        for (int off = 64; off > 0; off >>= 1) {
            if (d < off) red[d] += red[d + off];
            __syncthreads();
        }
        if (d == 0) s_att[k] = red[0] + ba[0];
        __syncthreads();
    }
    float mx = -3.4e38f;
    #pragma unroll
    for (int k = 0; k < NB_; ++k) mx = fmaxf(mx, s_att[k]);
    float al[NB_], sum = 0.0f;
    #pragma unroll
    for (int k = 0; k < NB_; ++k) { al[k] = __expf(s_att[k] - mx); sum += al[k]; }
    const float inv = 1.0f / sum;
    float agg = 0.0f;
    #pragma unroll
    for (int k = 0; k < NB_; ++k) agg += al[k] * inv * s_nb[k][d];
    X1[(size_t)node * D_ + d] = s_self[d] + agg;
}

// ---------------------------------------------------------------------------
// Sequential GRU. G = precomputed (x@Wih + bih) [B,S,384].
// grid = B/16 persistent blocks, block = 128 (4 waves). h kept in LDS,
// per-step gh = h @ Whh via WMMA (24 N-tiles, 6 per wave). Whh stays L2-hot.
// ---------------------------------------------------------------------------
__global__ void gru_seq(const float* __restrict__ G, const float* __restrict__ Whh,
                        const float* __restrict__ bhh, float* __restrict__ out) {
    __shared__ float sh[16 * 128];
    __shared__ float sgh[16 * 384];
    const int tid = threadIdx.x;
    const int b0 = blockIdx.x * 16;
    #pragma unroll
    for (int i = tid; i < 2048; i += 128) sh[i] = 0.0f;
    __syncthreads();

    const int wave = tid >> 5, lane = tid & 31;
    const int lm = lane & 15, hi = lane >> 4;

    for (int t = 0; t < S_; ++t) {
        // pull this step's input-gate rows toward the WGP while WMMAs run
        __builtin_prefetch(&G[((size_t)(b0 + (tid >> 3)) * S_ + t) * 384 + (tid & 7) * 48], 0, 0);

        #pragma unroll
        for (int jt = 0; jt < 6; ++jt) {
            const int n0 = (wave * 6 + jt) * 16;
            v8f acc = {};
            #pragma unroll 8
            for (int k0 = 0; k0 < 128; k0 += 4) {
                v2f a, bfr;
                a.x = sh[lm * 128 + k0 + 2 * hi + 0];
                a.y = sh[lm * 128 + k0 + 2 * hi + 1];
                bfr.x = Whh[(size_t)(k0 + 2 * hi + 0) * 384 + n0 + lm];
                bfr.y = Whh[(size_t)(k0 + 2 * hi + 1) * 384 + n0 + lm];
                acc = wmma4(a, bfr, acc);
            }
            #pragma unroll
            for (int r = 0; r < 8; ++r) sgh[(r + 8 * hi) * 384 + n0 + lm] = acc[r];
        }
        __syncthreads();

        const int d = tid;  // feature column
        #pragma unroll 4
        for (int row = 0; row < 16; ++row) {
            const int bi = b0 + row;
            const size_t gb = ((size_t)bi * S_ + t) * 384;
            const float ir = G[gb + d];
            const float iz = G[gb + 128 + d];
            const float in = G[gb + 256 + d];
            const float hr = sgh[row * 384 + d] + bhh[d];
            const float hz = sgh[row * 384 + 128 + d] + bhh[128 + d];
            const float hn = sgh[row * 384 + 256 + d] + bhh[256 + d];
            const float r = sigmoidf_(ir + hr);
            const float z = sigmoidf_(iz + hz);
            const float n = tanhf(in + r * hn);
            const float h2 = (1.0f - z) * n + z * sh[row * 128 + d];
            sh[row * 128 + d] = h2;
            out[((size_t)bi * S_ + t) * D_ + d] = h2;
        }
        __syncthreads();
    }
}

// ---------------------------------------------------------------------------
// local_ht = gru_out[b, clip(count_nonzero(h_iids[b])-1, 0, S-1)]
// ---------------------------------------------------------------------------
__global__ void last_hidden(const int* __restrict__ h_iids, const float* __restrict__ gru_out,
                            float* __restrict__ local_ht) {
    const int b = blockIdx.x, d = threadIdx.x;
    int cnt = 0;
    for (int s = 0; s < S_; ++s) cnt += (h_iids[b * S_ + s] != 0);
    int last = cnt - 1;
    last = last < 0 ? 0 : (last > S_ - 1 ? S_ - 1 : last);
    local_ht[(size_t)b * D_ + d] = gru_out[((size_t)b * S_ + last) * D_ + d];
}

// ---------------------------------------------------------------------------
// al[b,s] = <sigmoid(q1[b]+q2[b,s]), W3>;  global_ht[b] = sum_s al[b,s]*out[b,s]
// grid = B, block = 128
// ---------------------------------------------------------------------------
__global__ void alpha_global(const float* __restrict__ q1, const float* __restrict__ q2,
                             const float* __restrict__ W3, const float* __restrict__ gru_out,
                             float* __restrict__ global_ht) {
    const int b = blockIdx.x, d = threadIdx.x;
    __shared__ float red[128];
    __shared__ float s_al;
    const float q1d = q1[(size_t)b * D_ + d];
    const float w3d = W3[d];
    float acc = 0.0f;
    for (int s = 0; s < S_; ++s) {
        const float q2v = q2[((size_t)b * S_ + s) * D_ + d];
        red[d] = sigmoidf_(q1d + q2v) * w3d;
        __syncthreads();
        for (int off = 64; off > 0; off >>= 1) {
            if (d < off) red[d] += red[d + off];
            __syncthreads();
        }
        if (d == 0) s_al = red[0];
        __syncthreads();
        acc += s_al * gru_out[((size_t)b * S_ + s) * D_ + d];
        __syncthreads();
    }
    global_ht[(size_t)b * D_ + d] = acc;
}

// ---------------------------------------------------------------------------
// ght[b] = concat(local, global) @ Wtr + btr   (tiny: 8.4 MFLOP, plain VALU)
// ---------------------------------------------------------------------------
__global__ void transform_cat(const float* __restrict__ local_ht, const float* __restrict__ global_ht,
                              const float* __restrict__ Wtr, const float* __restrict__ btr,
                              float* __restrict__ ght) {
    const int b = blockIdx.x, d = threadIdx.x;
    __shared__ float sl[128], sg[128];
    sl[d] = local_ht[(size_t)b * D_ + d];
    sg[d] = global_ht[(size_t)b * D_ + d];
    __syncthreads();
    float acc = btr[d];
    #pragma unroll 8
    for (int k = 0; k < 128; ++k) {
        acc += sl[k] * Wtr[(size_t)k * D_ + d];
        acc += sg[k] * Wtr[(size_t)(k + 128) * D_ + d];
    }
    ght[(size_t)b * D_ + d] = acc;
}

// ---------------------------------------------------------------------------
// logits[b,i] = relu(<ght[b], item_emb[i]>).  A tile and a 64-row item tile
// staged into LDS via async b128 copies; B fragments built transposed from
// the LDS item tile.  OOB item rows are clamped: they feed only B columns
// whose outputs are never stored.  grid = (B/16, ceil(NI/64)), block = 128
// ---------------------------------------------------------------------------
__global__ void logits_kernel(const float* __restrict__ ght, const float* __restrict__ item,
                              float* __restrict__ out) {
    __shared__ __align__(16) float sA[16 * 128];
    __shared__ __align__(16) float sB[64 * 128];
    const int tid = threadIdx.x;
    const int m0 = blockIdx.x * 16;
    const int n0 = blockIdx.y * 64;

    {
        const unsigned ab = to_lds(sA);
        const float4* ght4 = (const float4*)ght + (size_t)m0 * 32;  // 32 float4/row
        #pragma unroll
        for (int i = tid; i < 512; i += 128)
            async_copy16(ab + (unsigned)i * 16u, ght4 + i);
    }
    {
        const unsigned bb = to_lds(sB);
        const float4* item4 = (const float4*)item;                  // 32 float4/row
        #pragma unroll
        for (int i = tid; i < 2048; i += 128) {
            int row = n0 + (i >> 5);
            if (row >= NI_) row = NI_ - 1;   // clamp: result column never stored
            async_copy16(bb + (unsigned)i * 16u, item4 + (size_t)row * 32 + (i & 31));
        }
    }
    wait_async();
    __syncthreads();

    const int wave = tid >> 5, lane = tid & 31;
    const int lm = lane & 15, hi = lane >> 4;
    const int nw = wave * 16;

    v8f acc = {};
    #pragma unroll 8
    for (int k0 = 0; k0 < 128; k0 += 4) {
        v2f a, b;
        a.x = sA[lm * 128 + k0 + 2 * hi + 0];
        a.y = sA[lm * 128 + k0 + 2 * hi + 1];
        b.x = sB[(nw + lm) * 128 + k0 + 2 * hi + 0];  // B[k][n] = item[n][k]
        b.y = sB[(nw + lm) * 128 + k0 + 2 * hi + 1];
        acc = wmma4(a, b, acc);
    }
    const int n = n0 + nw + lm;
    if (n < NI_) {
        #pragma unroll
        for (int r = 0; r < 8; ++r) {
            const int m = m0 + r + 8 * hi;
            const float v = acc[r];
            out[(size_t)m * NI_ + n] = v > 0.0f ? v : 0.0f;
        }
    }
}

// ---------------------------------------------------------------------------
extern "C" void kernel_launch(void* const* d_in, const int* in_sizes, int n_in,
                              void* d_out, int out_size, void* d_ws, size_t ws_size,
                              hipStream_t stream) {
    const int*   h_iids = (const int*)d_in[0];
    // d_in[1] = a_iids (unused, matches reference)
    const int*   adjE   = (const int*)d_in[2];
    const int*   adjR   = (const int*)d_in[3];
    const float* item   = (const float*)d_in[4];
    const float* relE   = (const float*)d_in[5];
    const float* Wa     = (const float*)d_in[6];
    const float* ba     = (const float*)d_in[7];
    const float* Wt     = (const float*)d_in[8];
    const float* bt     = (const float*)d_in[9];
    const float* Wih    = (const float*)d_in[10];
    const float* Whh    = (const float*)d_in[11];
    const float* bih    = (const float*)d_in[12];
    const float* bhh    = (const float*)d_in[13];
    const float* W1     = (const float*)d_in[14];
    const float* b1     = (const float*)d_in[15];
    const float* W2     = (const float*)d_in[16];
    const float* b2     = (const float*)d_in[17];
    const float* W3     = (const float*)d_in[18];
    const float* Wtr    = (const float*)d_in[19];
    const float* btr    = (const float*)d_in[20];
    float* out = (float*)d_out;

    // Workspace layout (two big ping-pong regions + small region).
    char* ws = (char*)d_ws;
    const size_t BIG = (size_t)B_ * S_ * NB_ * D_ * sizeof(float);   // 52.4 MB
    float* regA = (float*)(ws);                    // X0 -> X1 -> G -> q2
    float* regB = (float*)(ws + BIG);              // neib0 -> item_seq -> gru_out
    char*  smc  = ws + 2 * BIG;
    float* local_ht  = (float*)(smc + 0 * 131072);
    float* q1        = (float*)(smc + 1 * 131072);
    float* global_ht = (float*)(smc + 2 * 131072);
    float* ght       = (float*)(smc + 3 * 131072);

    // 1) level-0 fused gather + attention  -> X0 (regA) [102400,128]
    agg_level0<<<dim3(B_ * S_ * NB_), dim3(128), 0, stream>>>(h_iids, adjE, adjR, item, relE, Wa, ba, regA);
    // 2) neib0 = X0 @ Wt + bt  -> regB
    gemm_k128<<<dim3(B_ * S_ * NB_ / 16, 2), dim3(128), 0, stream>>>(regA, Wt, bt, regB, 128);
    // 3) level-1 attention (consumes neib0) -> X1 (regA) [12800,128]
    agg_level1<<<dim3(B_ * S_), dim3(128), 0, stream>>>(h_iids, adjR, item, relE, regB, Wa, ba, regA);
    // 4) item_seq = X1 @ Wt + bt -> regB
    gemm_k128<<<dim3(B_ * S_ / 16, 2), dim3(128), 0, stream>>>(regA, Wt, bt, regB, 128);
    // 5) G = item_seq @ Wih + bih -> regA [12800,384]
    gemm_k128<<<dim3(B_ * S_ / 16, 6), dim3(128), 0, stream>>>(regB, Wih, bih, regA, 384);
    // 6) GRU (sequential over t, recurrent GEMM in WMMA) -> gru_out (regB)
    gru_seq<<<dim3(B_ / 16), dim3(128), 0, stream>>>(regA, Whh, bhh, regB);
    // 7) local_ht
    last_hidden<<<dim3(B_), dim3(128), 0, stream>>>(h_iids, regB, local_ht);
    // 8) q1 = local_ht @ W1 + b1
    gemm_k128<<<dim3(B_ / 16, 2), dim3(128), 0, stream>>>(local_ht, W1, b1, q1, 128);
    // 9) q2 = gru_out @ W2 + b2 -> regA (G is dead)
    gemm_k128<<<dim3(B_ * S_ / 16, 2), dim3(128), 0, stream>>>(regB, W2, b2, regA, 128);
    // 10) attention pooling -> global_ht
    alpha_global<<<dim3(B_), dim3(128), 0, stream>>>(q1, regA, W3, regB, global_ht);
    // 11) ght = concat(local, global) @ Wtr + btr
    transform_cat<<<dim3(B_), dim3(128), 0, stream>>>(local_ht, global_ht, Wtr, btr, ght);
    // 12) logits = relu(ght @ item_emb^T) -> d_out [256, 100000]
    logits_kernel<<<dim3(B_ / 16, (NI_ + 63) / 64), dim3(128), 0, stream>>>(ght, item, out);
    (void)in_sizes; (void)n_in; (void)out_size; (void)ws_size;
}